// Proximity_Detector_with_Cats_37495064494412
// MI455X (gfx1250) — compile-verified
//
#include <hip/hip_runtime.h>

typedef __attribute__((ext_vector_type(2))) float v2f;
typedef __attribute__((ext_vector_type(8))) float v8f;

#define NB    1024
#define DIN   64
#define DHID  128
#define BN_EPS 1e-5f

// ---------------------------------------------------------------------------
// K1: A = x @ W1[:64,:] + b1   and   Bm = x @ W1[64:,:]
// fp32 WMMA 16x16x4, one wave per 16x16 C tile, K=64 in 16 steps.
// ---------------------------------------------------------------------------
__global__ __launch_bounds__(32)
void k1_gemm(const float* __restrict__ x, const float* __restrict__ W1,
             const float* __restrict__ b1, float* __restrict__ A,
             float* __restrict__ Bm)
{
    const int i0   = blockIdx.x * 16;
    const int j0   = blockIdx.y * 16;
    const int hlf  = blockIdx.z;            // 0 -> top half of W1 (A), 1 -> bottom (Bm)
    const int lane = threadIdx.x & 31;
    const int m    = lane & 15;
    const int kg   = (lane >> 4) << 1;      // 0 or 2 (K sub-group per half-wave)

    const float* __restrict__ W = W1 + hlf * DIN * DHID;
    float* __restrict__ C = hlf ? Bm : A;

    v8f acc = {0.f, 0.f, 0.f, 0.f, 0.f, 0.f, 0.f, 0.f};
    for (int k = 0; k < DIN; k += 4) {
        v2f a, b;
        // A-matrix 16x4: lane<16 -> (M=lane, K=k+0/1); lane>=16 -> (M=lane-16, K=k+2/3)
        a.x = x[(i0 + m) * DIN + (k + kg)];
        a.y = x[(i0 + m) * DIN + (k + kg + 1)];
        // B-matrix 4x16 (mirrored layout): row K, col N=m
        b.x = W[(k + kg) * DHID + (j0 + m)];
        b.y = W[(k + kg + 1) * DHID + (j0 + m)];
        acc = __builtin_amdgcn_wmma_f32_16x16x4_f32(false, a, false, b,
                                                    (short)0, acc, false, false);
    }

    const float bias  = (hlf == 0) ? b1[j0 + m] : 0.0f;
    const int   rbase = (lane < 16) ? 0 : 8;   // C/D: VGPR v -> M = v (+8 for hi lanes)
#pragma unroll
    for (int v = 0; v < 8; ++v) {
        C[(i0 + rbase + v) * DHID + (j0 + m)] = acc[v] + bias;
    }
}

// ---------------------------------------------------------------------------
// K2: per-feature partial sums S1=sum(h), S2=sum(h^2) over a 1024(i) x 256(j) slab.
// One block per (hidden unit h, j-chunk). Deterministic tree reduction.
// ---------------------------------------------------------------------------
__global__ __launch_bounds__(256)
void k2_stats(const float* __restrict__ A, const float* __restrict__ Bm,
              float* __restrict__ partials)
{
    const int h   = blockIdx.x;      // 0..127
    const int jc  = blockIdx.y;      // 0..3
    const int tid = threadIdx.x;

    __shared__ float bcol[256];
    __shared__ float red1[256];
    __shared__ float red2[256];

    bcol[tid] = Bm[(jc * 256 + tid) * DHID + h];

    const float a0 = A[(tid      ) * DHID + h];
    const float a1 = A[(tid + 256) * DHID + h];
    const float a2 = A[(tid + 512) * DHID + h];
    const float a3 = A[(tid + 768) * DHID + h];
    __syncthreads();

    float s1 = 0.f, s2 = 0.f;
    for (int j = 0; j < 256; ++j) {
        const float b  = bcol[j];
        const float r0 = fmaxf(a0 + b, 0.f);
        const float r1 = fmaxf(a1 + b, 0.f);
        const float r2 = fmaxf(a2 + b, 0.f);
        const float r3 = fmaxf(a3 + b, 0.f);
        s1 += (r0 + r1) + (r2 + r3);
        s2 = fmaf(r0, r0, s2); s2 = fmaf(r1, r1, s2);
        s2 = fmaf(r2, r2, s2); s2 = fmaf(r3, r3, s2);
    }
    red1[tid] = s1; red2[tid] = s2;
    __syncthreads();
    for (int s = 128; s > 0; s >>= 1) {
        if (tid < s) { red1[tid] += red1[tid + s]; red2[tid] += red2[tid + s]; }
        __syncthreads();
    }
    if (tid == 0) {
        partials[(h * 4 + jc) * 2 + 0] = red1[0];
        partials[(h * 4 + jc) * 2 + 1] = red2[0];
    }
}

// ---------------------------------------------------------------------------
// K3: finalize BN stats -> c[k] = W2*gamma*rsqrt(var+eps), scalar constant term.
// ---------------------------------------------------------------------------
__global__ __launch_bounds__(128)
void k3_final(const float* __restrict__ partials, const float* __restrict__ gamma,
              const float* __restrict__ beta, const float* __restrict__ W2,
              const float* __restrict__ b2, float* __restrict__ cvec,
              float* __restrict__ cterm)
{
    const int h = threadIdx.x;
    __shared__ float red[128];

    float S1 = 0.f, S2 = 0.f;
    for (int jc = 0; jc < 4; ++jc) {
        S1 += partials[(h * 4 + jc) * 2 + 0];
        S2 += partials[(h * 4 + jc) * 2 + 1];
    }
    const float invN = 1.0f / ((float)NB * (float)NB);
    const float mean = S1 * invN;
    const float var  = fmaxf(S2 * invN - mean * mean, 0.f);
    const float inv  = rsqrtf(var + BN_EPS);
    const float g    = gamma[h] * inv;
    cvec[h] = W2[h] * g;
    red[h]  = W2[h] * (beta[h] - mean * g);
    __syncthreads();
    for (int s = 64; s > 0; s >>= 1) {
        if (h < s) red[h] += red[h + s];
        __syncthreads();
    }
    if (h == 0) *cterm = red[0] + b2[0];
}

// ---------------------------------------------------------------------------
// K4: out(i,j) = sum_k c[k]*relu(A[i,k]+Bm[j,k]) + const.
// 64(i) x 32(j) tile per block; A/B tiles staged transposed in LDS ([h][i])
// so each thread reads contiguous float4/float2 fragments per k.
// ---------------------------------------------------------------------------
__global__ __launch_bounds__(256)
void k4_out(const float* __restrict__ A, const float* __restrict__ Bm,
            const float* __restrict__ cvec, const float* __restrict__ cterm,
            float* __restrict__ out)
{
    __shared__ float aT[DHID][64];   // 32 KB, [h][i_local]
    __shared__ float bT[DHID][32];   // 16 KB, [h][j_local]
    __shared__ float cs[DHID];

    const int i0  = blockIdx.x * 64;
    const int j0  = blockIdx.y * 32;
    const int tid = threadIdx.x;

    {   // stage A tile (64 rows x 128 h), transposed into LDS
        const int il = tid & 63;
        const int hq = tid >> 6;                       // 0..3 -> 32 h each
        const float4* src = (const float4*)(A + (i0 + il) * DHID + hq * 32);
#pragma unroll
        for (int q = 0; q < 8; ++q) {
            const float4 v = src[q];
            const int h = hq * 32 + q * 4;
            aT[h + 0][il] = v.x; aT[h + 1][il] = v.y;
            aT[h + 2][il] = v.z; aT[h + 3][il] = v.w;
        }
    }
    {   // stage B tile (32 rows x 128 h), transposed
        const int jl = tid & 31;
        const int hq = tid >> 5;                       // 0..7 -> 16 h each
        const float4* src = (const float4*)(Bm + (j0 + jl) * DHID + hq * 16);
#pragma unroll
        for (int q = 0; q < 4; ++q) {
            const float4 v = src[q];
            const int h = hq * 16 + q * 4;
            bT[h + 0][jl] = v.x; bT[h + 1][jl] = v.y;
            bT[h + 2][jl] = v.z; bT[h + 3][jl] = v.w;
        }
    }
    if (tid < DHID) cs[tid] = cvec[tid];
    __syncthreads();

    const int ib = (tid & 15) * 4;   // 4 i's per thread
    const int jb = (tid >> 4) * 2;   // 2 j's per thread
    float acc[4][2] = {};

    for (int h = 0; h < DHID; ++h) {
        const float4 av = *(const float4*)&aT[h][ib];
        const float  b0 = bT[h][jb];
        const float  b1v = bT[h][jb + 1];
        const float  ck = cs[h];
        acc[0][0] = fmaf(ck, fmaxf(av.x + b0, 0.f), acc[0][0]);
        acc[0][1] = fmaf(ck, fmaxf(av.x + b1v, 0.f), acc[0][1]);
        acc[1][0] = fmaf(ck, fmaxf(av.y + b0, 0.f), acc[1][0]);
        acc[1][1] = fmaf(ck, fmaxf(av.y + b1v, 0.f), acc[1][1]);
        acc[2][0] = fmaf(ck, fmaxf(av.z + b0, 0.f), acc[2][0]);
        acc[2][1] = fmaf(ck, fmaxf(av.z + b1v, 0.f), acc[2][1]);
        acc[3][0] = fmaf(ck, fmaxf(av.w + b0, 0.f), acc[3][0]);
        acc[3][1] = fmaf(ck, fmaxf(av.w + b1v, 0.f), acc[3][1]);
    }

    const float ct = *cterm;
#pragma unroll
    for (int ii = 0; ii < 4; ++ii) {
        float2 o;
        o.x = acc[ii][0] + ct;
        o.y = acc[ii][1] + ct;
        *(float2*)(out + (size_t)(i0 + ib + ii) * NB + (j0 + jb)) = o;
    }
}

// ---------------------------------------------------------------------------
extern "C" void kernel_launch(void* const* d_in, const int* in_sizes, int n_in,
                              void* d_out, int out_size, void* d_ws, size_t ws_size,
                              hipStream_t stream)
{
    const float* x     = (const float*)d_in[0];
    const float* W1    = (const float*)d_in[1];
    const float* b1    = (const float*)d_in[2];
    const float* gamma = (const float*)d_in[3];
    const float* beta  = (const float*)d_in[4];
    const float* W2    = (const float*)d_in[5];
    const float* b2    = (const float*)d_in[6];
    float* out = (float*)d_out;

    float* ws       = (float*)d_ws;
    float* A        = ws;                        // 1024*128
    float* Bm       = ws + NB * DHID;            // 1024*128
    float* partials = ws + 2 * NB * DHID;        // 128*4*2
    float* cvec     = partials + DHID * 4 * 2;   // 128
    float* cterm    = cvec + DHID;               // 1

    k1_gemm <<<dim3(NB / 16, DHID / 16, 2), 32,  0, stream>>>(x, W1, b1, A, Bm);
    k2_stats<<<dim3(DHID, 4),              256, 0, stream>>>(A, Bm, partials);
    k3_final<<<1,                          DHID, 0, stream>>>(partials, gamma, beta, W2, b2, cvec, cterm);
    k4_out  <<<dim3(NB / 64, NB / 32),     256, 0, stream>>>(A, Bm, cvec, cterm, out);
}